// AnchorLoss_335007450061
// MI455X (gfx1250) — compile-verified
//
#include <hip/hip_runtime.h>
#include <stdint.h>

#define NN 8
#define TT 64
#define AA 65536
#define CC 80
#define ALPHAF 0.25f
#define SL1_BETA (1.0f/9.0f)
#define EPSF 1e-6f
#define ND_BLOCKS 4096            // focal reduction blocks
#define NC_BLOCKS ((AA/256)*NN)   // 2048 code/box blocks

typedef float v2f __attribute__((ext_vector_type(2)));
typedef float v8f __attribute__((ext_vector_type(8)));

// ---- WMMA-based wave32 sum: A holds the 32 lane values (one per lane in VGPR0,
// zero in VGPR1), B = all-ones => D[m,n] = sum_k A[m,k]; summing a whole column
// (8 acc VGPRs) on lane L plus lane L^16 yields the total of all 32 inputs,
// independent of the exact A/B lane->element permutation.
__device__ __forceinline__ float wave_sum32(float v) {
  v2f a; a.x = v;    a.y = 0.0f;
  v2f b; b.x = 1.0f; b.y = 1.0f;
  v8f c = {};
  v8f d = __builtin_amdgcn_wmma_f32_16x16x4_f32(false, a, false, b, (short)0, c, false, false);
  float s = d[0] + d[1] + d[2] + d[3] + d[4] + d[5] + d[6] + d[7];
  s += __shfl_xor(s, 16, 32);
  return s;
}

// Valid on thread 0. Requires 256 threads, full EXEC.
__device__ __forceinline__ float block_sum256(float v, float* lds8) {
  float w = wave_sum32(v);
  if ((threadIdx.x & 31) == 0) lds8[threadIdx.x >> 5] = w;
  __syncthreads();
  float s = 0.0f;
  if (threadIdx.x == 0) {
#pragma unroll
    for (int i = 0; i < 8; ++i) s += lds8[i];
  }
  __syncthreads();
  return s;
}

__device__ __forceinline__ float focal_elem(float x, float y) {
  float e  = __expf(-fabsf(x));
  float ce = fmaxf(x, 0.0f) - x * y + __logf(1.0f + e);
  float r  = 1.0f / (1.0f + e);
  float p  = (x >= 0.0f) ? r : e * r;              // sigmoid(x), reusing e
  float pt = y * p + (1.0f - y) * (1.0f - p);
  float at = y * ALPHAF + (1.0f - y) * (1.0f - ALPHAF);
  float omt = 1.0f - pt;
  return at * omt * omt * ce;
}

__device__ __forceinline__ float sl1(float t, float p) {
  float d = fabsf(t - p);
  return (d < SL1_BETA) ? (0.5f * d * d / SL1_BETA) : (d - 0.5f * SL1_BETA);
}

// ---------------- kernel 0: init ----------------
__global__ void k_init(unsigned long long* gt_keys, int* pos_count) {
  int i = threadIdx.x;
  if (i < NN * TT) gt_keys[i] = 0ull;
  if (i < NN) pos_count[i] = 0;
}

// ---------------- kernel 1: per-anchor assignment ----------------
__global__ __launch_bounds__(256) void k_assign(
    const float4* __restrict__ bbox_true, const float4* __restrict__ anchors,
    float* __restrict__ max_iou, unsigned char* __restrict__ assigned,
    unsigned long long* __restrict__ gt_keys) {
  __shared__ float4 gbox[TT];
  __shared__ float  garea[TT];
  __shared__ int    gvalid[TT];
  __shared__ unsigned long long lkey[TT];
  const int n   = blockIdx.y;
  const int tid = threadIdx.x;
  if (tid < TT) {
    float4 g = bbox_true[n * TT + tid];
    gbox[tid]   = g;
    garea[tid]  = fmaxf(g.z - g.x, 0.0f) * fmaxf(g.w - g.y, 0.0f);
    gvalid[tid] = (g.x > 0.0f) || (g.y > 0.0f) || (g.z > 0.0f) || (g.w > 0.0f);
    lkey[tid]   = 0ull;
  }
  __syncthreads();
  const int a = blockIdx.x * 256 + tid;
  float4 an = anchors[a];
  float a2 = fmaxf(an.z - an.x, 0.0f) * fmaxf(an.w - an.y, 0.0f);
  float best = -1.0f; int best_t = 0;
  for (int t = 0; t < TT; ++t) {
    float iou;
    if (!gvalid[t]) {
      iou = -1.0f;
    } else {
      float4 g = gbox[t];
      float iw = fmaxf(fminf(g.z, an.z) - fmaxf(g.x, an.x), 0.0f);
      float ih = fmaxf(fminf(g.w, an.w) - fmaxf(g.y, an.y), 0.0f);
      float inter = iw * ih;
      float uni = garea[t] + a2 - inter;
      iou = inter / fmaxf(uni, 1e-10f);
      if (iou > 0.0f) {
        unsigned long long key =
            ((unsigned long long)__float_as_uint(iou) << 32) |
            (unsigned long long)(0xFFFFFFFFu - (unsigned)a);   // tie -> lowest anchor
        atomicMax(&lkey[t], key);
      }
    }
    if (iou > best) { best = iou; best_t = t; }                // tie -> first t
  }
  max_iou[n * AA + a]  = best;
  assigned[n * AA + a] = (unsigned char)best_t;
  __syncthreads();
  if (tid < TT && lkey[tid] != 0ull) atomicMax(&gt_keys[n * TT + tid], lkey[tid]);
}

// ---------------- kernel 2: decode gt-best + labels ----------------
__global__ void k_decode(const unsigned long long* __restrict__ gt_keys,
                         const float* __restrict__ y_true,
                         int* __restrict__ lq_anchor, int* __restrict__ labels) {
  int i = threadIdx.x;   // 512 = N*T
  if (i >= NN * TT) return;
  unsigned long long key = gt_keys[i];
  float giou = __uint_as_float((unsigned)(key >> 32));
  lq_anchor[i] = (giou > 0.0f) ? (int)(0xFFFFFFFFu - (unsigned)(key & 0xFFFFFFFFu)) : -1;
  const float* row = y_true + (size_t)i * CC;
  int lab = 0;
  for (int c = 0; c < CC; ++c) { if (row[c] > 0.5f) { lab = c; break; } }
  labels[i] = lab;
}

// ---------------- kernel 3: final state/code + smooth-L1 ----------------
__global__ __launch_bounds__(256) void k_codebox(
    const float* __restrict__ max_iou, const unsigned char* __restrict__ assigned,
    const int* __restrict__ lq_anchor, const int* __restrict__ labels,
    const float4* __restrict__ anchors, const float4* __restrict__ bbox_true,
    const float4* __restrict__ bbox_pred,
    unsigned char* __restrict__ code, int* __restrict__ pos_count,
    float* __restrict__ box_partials) {
  __shared__ int   lqa[TT];
  __shared__ float lds8[8];
  const int n   = blockIdx.y;
  const int tid = threadIdx.x;
  if (tid < TT) lqa[tid] = lq_anchor[n * TT + tid];
  __syncthreads();
  const int a = blockIdx.x * 256 + tid;
  float mi = max_iou[n * AA + a];
  int asg  = assigned[n * AA + a];
  int lq_t = -1;
#pragma unroll
  for (int t = 0; t < TT; ++t) if (lqa[t] == a) lq_t = t;   // max-t wins
  bool pos = (mi >= 0.5f);
  if (lq_t >= 0) { pos = true; asg = lq_t; }
  bool neg = (mi < 0.4f) && !pos;
  unsigned char cd = pos ? (unsigned char)(2 + labels[n * TT + asg]) : (neg ? 1 : 0);
  code[n * AA + a] = cd;
  float bsum = 0.0f;
  if (pos) {
    float4 an = anchors[a];
    float4 g  = bbox_true[n * TT + asg];
    float4 bp = bbox_pred[n * AA + a];
    float wa = fmaxf(an.z - an.x, EPSF), ha = fmaxf(an.w - an.y, EPSF);
    float cxa = an.x + 0.5f * wa, cya = an.y + 0.5f * ha;
    float wt = fmaxf(g.z - g.x, EPSF), ht = fmaxf(g.w - g.y, EPSF);
    float cxt = g.x + 0.5f * (g.z - g.x), cyt = g.y + 0.5f * (g.w - g.y);
    bsum = sl1((cxt - cxa) / wa, bp.x) + sl1((cyt - cya) / ha, bp.y) +
           sl1(__logf(wt / wa), bp.z) + sl1(__logf(ht / ha), bp.w);
  }
  unsigned long long bal = __ballot(pos);
  if ((tid & 31) == 0) atomicAdd(&pos_count[n], (int)__popcll(bal));
  float tot = block_sum256(bsum, lds8);
  if (tid == 0) box_partials[blockIdx.y * (AA / 256) + blockIdx.x] = tot;
}

// ---------------- kernel 4: focal loss streaming pass (B128 loads) ----------------
__global__ __launch_bounds__(256) void k_focal(
    const float4* __restrict__ y_pred4, const unsigned char* __restrict__ code,
    float* __restrict__ cls_partials) {
  __shared__ float lds8[8];
  const unsigned total4 = (unsigned)NN * AA * (CC / 4);   // 10,485,760 float4s
  const unsigned stride = (unsigned)ND_BLOCKS * 256;      // 1,048,576 -> exactly 10 iters
  float acc = 0.0f;
  for (unsigned i4 = blockIdx.x * 256 + threadIdx.x; i4 < total4; i4 += stride) {
    unsigned na = i4 / (CC / 4);                // 20 float4 per anchor row
    unsigned c0 = (i4 - na * (CC / 4)) * 4;     // first channel of this float4
    unsigned char cd = code[na];
    if (cd != 0) {
      float4 x = y_pred4[i4];
      unsigned ulab = (unsigned)((int)cd - 2);  // 0xFFFFFFFF for negatives
      acc += focal_elem(x.x, (c0 + 0u == ulab) ? 1.0f : 0.0f);
      acc += focal_elem(x.y, (c0 + 1u == ulab) ? 1.0f : 0.0f);
      acc += focal_elem(x.z, (c0 + 2u == ulab) ? 1.0f : 0.0f);
      acc += focal_elem(x.w, (c0 + 3u == ulab) ? 1.0f : 0.0f);
    }
  }
  float tot = block_sum256(acc, lds8);
  if (threadIdx.x == 0) cls_partials[blockIdx.x] = tot;
}

// ---------------- kernel 5: final deterministic reduction ----------------
__global__ __launch_bounds__(256) void k_final(
    const float* __restrict__ cls_partials, const float* __restrict__ box_partials,
    const int* __restrict__ pos_count, float* __restrict__ out) {
  __shared__ float lds8[8];
  float cs = 0.0f, bs = 0.0f;
  for (int i = threadIdx.x; i < ND_BLOCKS; i += 256) cs += cls_partials[i];
  for (int i = threadIdx.x; i < NC_BLOCKS; i += 256) bs += box_partials[i];
  float ct = block_sum256(cs, lds8);
  float bt = block_sum256(bs, lds8);
  if (threadIdx.x == 0) {
    float af = 0.0f;
    for (int n = 0; n < NN; ++n) af += fmaxf((float)pos_count[n], 1.0f);
    float c = ct / af, b = bt / af;
    if (isnan(c) || isinf(c)) c = 0.0f;
    if (isnan(b) || isinf(b)) b = 0.0f;
    out[0] = c; out[1] = b;
  }
}

extern "C" void kernel_launch(void* const* d_in, const int* in_sizes, int n_in,
                              void* d_out, int out_size, void* d_ws, size_t ws_size,
                              hipStream_t stream) {
  (void)in_sizes; (void)n_in; (void)out_size; (void)ws_size;
  const float* y_true    = (const float*)d_in[0];   // (N,T,C)
  const float* bbox_true = (const float*)d_in[1];   // (N,T,4)
  const float* y_pred    = (const float*)d_in[2];   // (N,A,C)
  const float* bbox_pred = (const float*)d_in[3];   // (N,A,4)
  const float* anchors   = (const float*)d_in[4];   // (A,4)

  char* ws = (char*)d_ws;
  unsigned long long* gt_keys = (unsigned long long*)(ws);            // 4096 B
  int* labels    = (int*)(ws + 4096);                                 // 2048 B
  int* lq_anchor = (int*)(ws + 6144);                                 // 2048 B
  int* pos_count = (int*)(ws + 8192);                                 // 32 B
  float* max_iou = (float*)(ws + 8448);                               // 2 MB
  unsigned char* assigned = (unsigned char*)(ws + 8448 + (size_t)NN * AA * 4);       // 512 KB
  unsigned char* code     = assigned + (size_t)NN * AA;                              // 512 KB
  float* cls_partials = (float*)(ws + 8448 + (size_t)NN * AA * 4 + 2 * (size_t)NN * AA);
  float* box_partials = cls_partials + ND_BLOCKS;
  float* out = (float*)d_out;

  k_init<<<1, 512, 0, stream>>>(gt_keys, pos_count);
  k_assign<<<dim3(AA / 256, NN), 256, 0, stream>>>(
      (const float4*)bbox_true, (const float4*)anchors, max_iou, assigned, gt_keys);
  k_decode<<<1, 512, 0, stream>>>(gt_keys, y_true, lq_anchor, labels);
  k_codebox<<<dim3(AA / 256, NN), 256, 0, stream>>>(
      max_iou, assigned, lq_anchor, labels, (const float4*)anchors,
      (const float4*)bbox_true, (const float4*)bbox_pred, code, pos_count, box_partials);
  k_focal<<<ND_BLOCKS, 256, 0, stream>>>((const float4*)y_pred, code, cls_partials);
  k_final<<<1, 256, 0, stream>>>(cls_partials, box_partials, pos_count, out);
}